// HyperDiffusion_25013889532002
// MI455X (gfx1250) — compile-verified
//
#include <hip/hip_runtime.h>
#include <hip/hip_bf16.h>

// HyperDiffusion forward for MI455X (gfx1250, wave32).
//
// Strategy: the op is a sparse incidence scatter/gather (NNZ=1.6M, C=128).
// Atomic-op count dominates a naive scatter (410M fp32 atomics), so we build
// CSR on the fly (histogram + scan + permutation fill: ~6.4M int atomics) and
// run both diffusion passes as register-accumulating GATHER kernels with a
// single b128 store per output row. All feature tables are L2-resident
// (X 51MB, node 51MB, edge 10MB << 192MB L2), so the gather passes run at
// L2 bandwidth with global_prefetch_b8 hiding the random-row latency.

#define C_DIM 128
#define LANES 32
#define SCAN_T 1024

// ---------------------------------------------------------------- zero fills
__global__ __launch_bounds__(256) void hd_zero4_kernel(float4* __restrict__ p, int n4) {
    int i = blockIdx.x * blockDim.x + threadIdx.x;
    if (i < n4) p[i] = make_float4(0.f, 0.f, 0.f, 0.f);
}
__global__ __launch_bounds__(256) void hd_zero1_kernel(int* __restrict__ p, int n) {
    int i = blockIdx.x * blockDim.x + threadIdx.x;
    if (i < n) p[i] = 0;
}

// ---------------------------------------------------------------- histogram
__global__ __launch_bounds__(256) void hd_hist_kernel(const int* __restrict__ vid,
                                                      const int* __restrict__ eid,
                                                      int* __restrict__ cnt_v,
                                                      int* __restrict__ cnt_e,
                                                      int nnz) {
    int i = blockIdx.x * blockDim.x + threadIdx.x;
    if (i < nnz) {
        atomicAdd(&cnt_v[vid[i]], 1);   // global_atomic_add_u32 (no return)
        atomicAdd(&cnt_e[eid[i]], 1);
    }
}

// ---------------------------------------------------------------- exclusive scan
// Single workgroup (1024 threads = 32 waves), chunked Hillis-Steele in LDS.
// Writes exclusive offsets to `off` (n+1 entries) and a cursor copy to `cur`.
__global__ __launch_bounds__(SCAN_T) void hd_scan_kernel(const int* __restrict__ counts,
                                                         int* __restrict__ off,
                                                         int* __restrict__ cur,
                                                         int n) {
    __shared__ int buf[SCAN_T];
    __shared__ int carry;
    const int tid = threadIdx.x;
    if (tid == 0) carry = 0;
    __syncthreads();

    for (int base = 0; base < n; base += SCAN_T) {
        const int i = base + tid;
        const int x = (i < n) ? counts[i] : 0;
        buf[tid] = x;
        __syncthreads();
        for (int d = 1; d < SCAN_T; d <<= 1) {
            int t = (tid >= d) ? buf[tid - d] : 0;
            __syncthreads();
            buf[tid] += t;
            __syncthreads();
        }
        const int incl = buf[tid];
        const int c = carry;                 // everyone reads old carry
        __syncthreads();
        if (tid == 0) carry = c + buf[SCAN_T - 1];
        const int excl = c + incl - x;
        if (i < n) { off[i] = excl; cur[i] = excl; }
        __syncthreads();                     // carry visible; buf reusable
    }
    if (tid == 0) off[n] = carry;
}

// ---------------------------------------------------------------- inverse degree
__global__ __launch_bounds__(256) void hd_invdeg_kernel(const int* __restrict__ off,
                                                        float* __restrict__ inv, int n) {
    int i = blockIdx.x * blockDim.x + threadIdx.x;
    if (i < n) {
        int c = off[i + 1] - off[i];
        inv[i] = (c > 0) ? (1.0f / (float)c) : 0.f;
    }
}

// ---------------------------------------------------------------- CSR fill
__global__ __launch_bounds__(256) void hd_fill_kernel(const int* __restrict__ vid,
                                                      const int* __restrict__ eid,
                                                      int* __restrict__ cur_e,
                                                      int* __restrict__ cur_v,
                                                      int* __restrict__ v_by_e,
                                                      int* __restrict__ e_by_v,
                                                      int nnz) {
    int i = blockIdx.x * blockDim.x + threadIdx.x;
    if (i < nnz) {
        const int v = vid[i];
        const int e = eid[i];
        int p = atomicAdd(&cur_e[e], 1);    // global_atomic_add_u32 RTN
        v_by_e[p] = v;
        int q = atomicAdd(&cur_v[v], 1);
        e_by_v[q] = e;
    }
}

// ---------------------------------------------------------------- gather-sum
// One wave per output segment; lane l owns channels [4l,4l+4) as float4.
// acc in registers, 2-way unrolled with double prefetch (global_prefetch_b8),
// single coalesced b128 store per row -> zero fp32 atomics.
__global__ __launch_bounds__(256) void hd_gather_kernel(const float* __restrict__ src,
                                                        const int* __restrict__ off,
                                                        const int* __restrict__ memb,
                                                        const float* __restrict__ inv_memb,
                                                        float* __restrict__ dst,
                                                        int nseg) {
    const int lane   = threadIdx.x & (LANES - 1);
    const int wave   = blockIdx.x * (blockDim.x >> 5) + (threadIdx.x >> 5);
    const int nwaves = gridDim.x * (blockDim.x >> 5);

    for (int s = wave; s < nseg; s += nwaves) {
        const int b = off[s];
        const int e = off[s + 1];
        float4 a0 = make_float4(0.f, 0.f, 0.f, 0.f);
        float4 a1 = make_float4(0.f, 0.f, 0.f, 0.f);

        int j = b;
        for (; j + 1 < e; j += 2) {
            const int m0 = memb[j];
            const int m1 = memb[j + 1];
            if (j + 3 < e) {   // prefetch two rows ahead
                __builtin_prefetch(&src[(size_t)memb[j + 2] * C_DIM + lane * 4], 0, 3);
                __builtin_prefetch(&src[(size_t)memb[j + 3] * C_DIM + lane * 4], 0, 3);
            }
            const float w0 = inv_memb[m0];
            const float w1 = inv_memb[m1];
            const float4 x0 = *(const float4*)(&src[(size_t)m0 * C_DIM + lane * 4]);
            const float4 x1 = *(const float4*)(&src[(size_t)m1 * C_DIM + lane * 4]);
            a0.x = fmaf(x0.x, w0, a0.x); a0.y = fmaf(x0.y, w0, a0.y);
            a0.z = fmaf(x0.z, w0, a0.z); a0.w = fmaf(x0.w, w0, a0.w);
            a1.x = fmaf(x1.x, w1, a1.x); a1.y = fmaf(x1.y, w1, a1.y);
            a1.z = fmaf(x1.z, w1, a1.z); a1.w = fmaf(x1.w, w1, a1.w);
        }
        if (j < e) {
            const int m = memb[j];
            const float w = inv_memb[m];
            const float4 x = *(const float4*)(&src[(size_t)m * C_DIM + lane * 4]);
            a0.x = fmaf(x.x, w, a0.x); a0.y = fmaf(x.y, w, a0.y);
            a0.z = fmaf(x.z, w, a0.z); a0.w = fmaf(x.w, w, a0.w);
        }
        float4 r = make_float4(a0.x + a1.x, a0.y + a1.y, a0.z + a1.z, a0.w + a1.w);
        *(float4*)(&dst[(size_t)s * C_DIM + lane * 4]) = r;
    }
}

// ================================================================ fallback
// Atomic-scatter path used only if the workspace is too small for CSR.
__global__ __launch_bounds__(256) void hd_degree_f_kernel(const int* __restrict__ vid,
                                                          const int* __restrict__ eid,
                                                          float* __restrict__ deg_v,
                                                          float* __restrict__ deg_e,
                                                          int nnz) {
    int i = blockIdx.x * blockDim.x + threadIdx.x;
    if (i < nnz) {
        unsafeAtomicAdd(&deg_v[vid[i]], 1.0f);
        unsafeAtomicAdd(&deg_e[eid[i]], 1.0f);
    }
}
__global__ __launch_bounds__(256) void hd_invf_kernel(float* __restrict__ deg, int n) {
    int i = blockIdx.x * blockDim.x + threadIdx.x;
    if (i < n) { float d = deg[i]; deg[i] = (d > 0.f) ? (1.0f / d) : 0.f; }
}
__global__ __launch_bounds__(256) void hd_scatter_kernel(const float* __restrict__ src,
                                                         const int* __restrict__ gidx,
                                                         const int* __restrict__ sidx,
                                                         const float* __restrict__ inv_deg,
                                                         float* __restrict__ dst,
                                                         int nnz) {
    const int lane   = threadIdx.x & (LANES - 1);
    const int wave   = blockIdx.x * (blockDim.x >> 5) + (threadIdx.x >> 5);
    const int nwaves = gridDim.x * (blockDim.x >> 5);
    for (int i = wave; i < nnz; i += nwaves) {
        const int g = gidx[i];
        const int s = sidx[i];
        const int inext = i + nwaves;
        if (inext < nnz)
            __builtin_prefetch(&src[(size_t)gidx[inext] * C_DIM + lane * 4], 0, 3);
        const float w = inv_deg[g];
        const float4 x = *(const float4*)(&src[(size_t)g * C_DIM + lane * 4]);
        float* d = &dst[(size_t)s * C_DIM + lane * 4];
        unsafeAtomicAdd(d + 0, x.x * w);
        unsafeAtomicAdd(d + 1, x.y * w);
        unsafeAtomicAdd(d + 2, x.z * w);
        unsafeAtomicAdd(d + 3, x.w * w);
    }
}

// ---------------------------------------------------------------- launcher
extern "C" void kernel_launch(void* const* d_in, const int* in_sizes, int n_in,
                              void* d_out, int out_size, void* d_ws, size_t ws_size,
                              hipStream_t stream) {
    const float* X          = (const float*)d_in[0];
    const int*   vertex_ids = (const int*)d_in[1];
    const int*   edge_ids   = (const int*)d_in[2];

    const int nnz = in_sizes[1];
    const int N   = in_sizes[0] / C_DIM;
    const int E   = (out_size - in_sizes[0]) / C_DIM;

    float* node_feat = (float*)d_out;                     // [N, C]
    float* edge_feat = (float*)d_out + (size_t)N * C_DIM; // [E, C]

    const int BLK = 256;
    const size_t need = (size_t)((E + 1) + (N + 1) + E + N + 2 * (size_t)nnz + E + N) * 4;

    if (ws_size >= need) {
        // ---------------- CSR gather path ----------------
        int* off_e  = (int*)d_ws;             // E+1
        int* off_v  = off_e + (E + 1);        // N+1
        int* cur_e  = off_v + (N + 1);        // E   (counts, then cursors)
        int* cur_v  = cur_e + E;              // N
        int* v_by_e = cur_v + N;              // nnz
        int* e_by_v = v_by_e + nnz;           // nnz
        float* inv_e = (float*)(e_by_v + nnz);// E
        float* inv_v = inv_e + E;             // N

        // 1) zero counts (cur_e|cur_v contiguous)
        hd_zero1_kernel<<<(E + N + BLK - 1) / BLK, BLK, 0, stream>>>(cur_e, E + N);
        // 2) histogram
        hd_hist_kernel<<<(nnz + BLK - 1) / BLK, BLK, 0, stream>>>(vertex_ids, edge_ids,
                                                                  cur_v, cur_e, nnz);
        // 3) exclusive scans -> offsets + cursor copies
        hd_scan_kernel<<<1, SCAN_T, 0, stream>>>(cur_e, off_e, cur_e, E);
        hd_scan_kernel<<<1, SCAN_T, 0, stream>>>(cur_v, off_v, cur_v, N);
        // 4) inverse degrees
        hd_invdeg_kernel<<<(E + BLK - 1) / BLK, BLK, 0, stream>>>(off_e, inv_e, E);
        hd_invdeg_kernel<<<(N + BLK - 1) / BLK, BLK, 0, stream>>>(off_v, inv_v, N);
        // 5) permutation fill
        hd_fill_kernel<<<(nnz + BLK - 1) / BLK, BLK, 0, stream>>>(vertex_ids, edge_ids,
                                                                  cur_e, cur_v,
                                                                  v_by_e, e_by_v, nnz);
        // 6) v2e gather: edge_feat[e] = sum_v X[v] * inv_deg_v[v]
        {
            int blocks = (E + (BLK / LANES) - 1) / (BLK / LANES);
            if (blocks > 16384) blocks = 16384;
            hd_gather_kernel<<<blocks, BLK, 0, stream>>>(X, off_e, v_by_e, inv_v,
                                                         edge_feat, E);
        }
        // 7) e2v gather: node_feat[v] = sum_e edge_feat[e] * inv_deg_e[e]
        {
            int blocks = (N + (BLK / LANES) - 1) / (BLK / LANES);
            if (blocks > 16384) blocks = 16384;
            hd_gather_kernel<<<blocks, BLK, 0, stream>>>(edge_feat, off_v, e_by_v, inv_e,
                                                         node_feat, N);
        }
        // every output row written exactly once -> no d_out zeroing needed
    } else {
        // ---------------- atomic fallback ----------------
        float* deg_v = (float*)d_ws;
        float* deg_e = deg_v + N;
        {
            int n4 = out_size / 4;
            hd_zero4_kernel<<<(n4 + BLK - 1) / BLK, BLK, 0, stream>>>((float4*)d_out, n4);
        }
        hd_zero1_kernel<<<(N + E + BLK - 1) / BLK, BLK, 0, stream>>>((int*)d_ws, N + E);
        hd_degree_f_kernel<<<(nnz + BLK - 1) / BLK, BLK, 0, stream>>>(vertex_ids, edge_ids,
                                                                      deg_v, deg_e, nnz);
        hd_invf_kernel<<<(N + E + BLK - 1) / BLK, BLK, 0, stream>>>(deg_v, N + E);
        int blocks = (nnz + (BLK / LANES) - 1) / (BLK / LANES);
        if (blocks > 16384) blocks = 16384;
        hd_scatter_kernel<<<blocks, BLK, 0, stream>>>(X, vertex_ids, edge_ids,
                                                      deg_v, edge_feat, nnz);
        hd_scatter_kernel<<<blocks, BLK, 0, stream>>>(edge_feat, edge_ids, vertex_ids,
                                                      deg_e, node_feat, nnz);
    }
}